// PermutoEncCat_71262097375540
// MI455X (gfx1250) — compile-verified
//
#include <hip/hip_runtime.h>
#include <hip/hip_bf16.h>

typedef __attribute__((ext_vector_type(2))) float v2f;
typedef __attribute__((ext_vector_type(8))) float v8f;

#define LVLS 16
#define TSZ  (1 << 19)
#define NPTS 262144
#define BLK  64
#define NWAVE (BLK / 32)

__device__ __forceinline__ v8f wmma_f32(v2f a, v2f b, v8f c) {
  // D = A(16x4,f32) * B(4x16,f32) + C(16x16,f32)
  return __builtin_amdgcn_wmma_f32_16x16x4_f32(false, a, false, b, (short)0, c,
                                               false, false);
}

// SF constants: 1/sqrt((d+1)(d+2))
#define SF0 0.70710678f
#define SF1 0.40824829f
#define SF2 0.28867513f
#define SF3 0.22360680f

// One permutohedral level: produce hash indices, ranks, barycentric weights.
__device__ __forceinline__ void permuto_level(const float pos[4], float scale,
                                              int idx[5], int rank[5],
                                              float bary[5]) {
  const float c0 = pos[0] * scale * SF0;
  const float c1 = pos[1] * scale * SF1;
  const float c2 = pos[2] * scale * SF2;
  const float c3 = pos[3] * scale * SF3;
  float el[5];
  el[0] = c0 + c1 + c2 + c3;
  el[1] = -c0 + c1 + c2 + c3;
  el[2] = -2.f * c1 + c2 + c3;
  el[3] = -3.f * c2 + c3;
  el[4] = -4.f * c3;

  float rem0[5];
  float sumr = 0.f;
#pragma unroll
  for (int v = 0; v < 5; ++v) {
    float q = el[v] * 0.2f;
    float up = ceilf(q) * 5.f;
    float dn = floorf(q) * 5.f;
    float r = ((up - el[v]) < (el[v] - dn)) ? up : dn;
    rem0[v] = r;
    sumr += r;
  }
  const int s = (int)rintf(sumr * 0.2f);

  float diff[5];
#pragma unroll
  for (int v = 0; v < 5; ++v) diff[v] = el[v] - rem0[v];
#pragma unroll
  for (int i = 0; i < 5; ++i) {
    int rk = 0;
#pragma unroll
    for (int j = 0; j < 5; ++j) {
      if (j > i) rk += (diff[i] < diff[j]) ? 1 : 0;
      if (j < i) rk += (diff[i] <= diff[j]) ? 1 : 0;
    }
    rank[i] = rk + s;
  }
#pragma unroll
  for (int v = 0; v < 5; ++v) {
    if (rank[v] < 0) {
      rank[v] += 5;
      rem0[v] += 5.f;
    } else if (rank[v] > 4) {
      rank[v] -= 5;
      rem0[v] -= 5.f;
    }
  }

  // bary[b] = sum_v delta_v * (one_hot(4-rank_v) - one_hot(5-rank_v)); static
  // indices only (select-reduction) so everything stays in VGPRs.
  float b6[6];
#pragma unroll
  for (int bi = 0; bi < 6; ++bi) {
    float acc = 0.f;
#pragma unroll
    for (int v = 0; v < 5; ++v) {
      const float d = (el[v] - rem0[v]) * 0.2f;
      acc += (4 - rank[v] == bi) ? d : 0.f;
      acc -= (5 - rank[v] == bi) ? d : 0.f;
    }
    b6[bi] = acc;
  }
  b6[0] += 1.f + b6[5];
#pragma unroll
  for (int k = 0; k < 5; ++k) bary[k] = b6[k];

  int r0[4];
#pragma unroll
  for (int d = 0; d < 4; ++d) r0[d] = (int)rem0[d];
  const unsigned P1 = 2654435761u, P2 = 805459861u, P3 = 3674653429u;
#pragma unroll
  for (int k = 0; k < 5; ++k) {
    unsigned h = (unsigned)(r0[0] + k - ((rank[0] > 4 - k) ? 5 : 0));
    h ^= (unsigned)(r0[1] + k - ((rank[1] > 4 - k) ? 5 : 0)) * P1;
    h ^= (unsigned)(r0[2] + k - ((rank[2] > 4 - k) ? 5 : 0)) * P2;
    h ^= (unsigned)(r0[3] + k - ((rank[3] > 4 - k) ? 5 : 0)) * P3;
    idx[k] = (int)(h & (unsigned)(TSZ - 1));
  }
}

__global__ __launch_bounds__(BLK) void permuto_sdf_kernel(
    const float* __restrict__ x, const float* __restrict__ z,
    const float* __restrict__ table, const float* __restrict__ W1,
    const float* __restrict__ b1, const float* __restrict__ W2,
    const float* __restrict__ b2, const float* __restrict__ W3,
    const float* __restrict__ b3, float* __restrict__ out) {
  __shared__ float sW1[32 * 64];
  __shared__ float sW2[64 * 64];
  __shared__ float sW3[64];
  __shared__ float sB1[64];
  __shared__ float sB2[64];
  __shared__ float sB3[1];
  __shared__ float sH0[BLK][33];          // features, later grad-features
  __shared__ float sH1[NWAVE][16 * 65];   // per-wave tile h1 / g1
  __shared__ float sH2[NWAVE][16 * 65];   // per-wave tile h2

  const int tid = threadIdx.x;
  for (int i = tid; i < 32 * 64; i += BLK) sW1[i] = W1[i];
  for (int i = tid; i < 64 * 64; i += BLK) sW2[i] = W2[i];
  if (tid < 64) {
    sW3[tid] = W3[tid];
    sB1[tid] = b1[tid];
    sB2[tid] = b2[tid];
  }
  if (tid == 0) sB3[0] = b3[0];

  const int gp = blockIdx.x * BLK + tid;
  float pos[4];
  pos[0] = x[gp * 3 + 0] * 0.5f + 0.5f;
  pos[1] = x[gp * 3 + 1] * 0.5f + 0.5f;
  pos[2] = x[gp * 3 + 2] * 0.5f + 0.5f;
  pos[3] = z[gp];

  // ---------------- Phase 1: permutohedral encoding (forward) --------------
  // unroll 2: keep 10 random L2 gathers in flight per thread and overlap one
  // level's rank/hash VALU work with the next level's table loads.
#pragma unroll 2
  for (int l = 0; l < LVLS; ++l) {
    const float scale = exp2f(fmaf((float)l, 7.0f / 15.0f, 4.0f));
    int idx[5], rank[5];
    float bary[5];
    permuto_level(pos, scale, idx, rank, bary);
    const float2* tb = (const float2*)table + (size_t)l * TSZ;
    float f0 = 0.f, f1 = 0.f;
#pragma unroll
    for (int k = 0; k < 5; ++k) {
      const float2 v = tb[idx[k]];
      f0 = fmaf(bary[k], v.x, f0);
      f1 = fmaf(bary[k], v.y, f1);
    }
    sH0[tid][2 * l] = f0;
    sH0[tid][2 * l + 1] = f1;
  }
  __syncthreads();

  // ---------------- Phase 2: fused MLP fwd + bwd via fp32 WMMA -------------
  const int w = tid >> 5;
  const int lane = tid & 31;
  const int mn = lane & 15;          // A row / B-C column within tile
  const int kb = (lane >> 4) << 1;   // K sub-offset: 0 or 2
  const int mh = (lane >> 4) << 3;   // C/D row offset: 0 or 8
  float* H1 = sH1[w];
  float* H2 = sH2[w];

#pragma unroll 1
  for (int t = 0; t < 2; ++t) {
    const int pb = w * 32 + t * 16;  // local point base of this 16-row tile

    // Layer 1: h1 = relu(h0 @ W1 + b1)   (16x32)x(32x64)
#pragma unroll 1
    for (int n0 = 0; n0 < 4; ++n0) {
      v8f acc = {};
#pragma unroll
      for (int kk = 0; kk < 32; kk += 4) {
        v2f a, b;
        a.x = sH0[pb + mn][kk + kb + 0];
        a.y = sH0[pb + mn][kk + kb + 1];
        b.x = sW1[(kk + kb + 0) * 64 + n0 * 16 + mn];
        b.y = sW1[(kk + kb + 1) * 64 + n0 * 16 + mn];
        acc = wmma_f32(a, b, acc);
      }
      const int col = n0 * 16 + mn;
      const float bias = sB1[col];
#pragma unroll
      for (int j = 0; j < 8; ++j)
        H1[(j + mh) * 65 + col] = fmaxf(acc[j] + bias, 0.f);
    }

    // Layer 2: h2 = relu(h1 @ W2 + b2)   (16x64)x(64x64)
#pragma unroll 1
    for (int n0 = 0; n0 < 4; ++n0) {
      v8f acc = {};
#pragma unroll
      for (int kk = 0; kk < 64; kk += 4) {
        v2f a, b;
        a.x = H1[mn * 65 + kk + kb + 0];
        a.y = H1[mn * 65 + kk + kb + 1];
        b.x = sW2[(kk + kb + 0) * 64 + n0 * 16 + mn];
        b.y = sW2[(kk + kb + 1) * 64 + n0 * 16 + mn];
        acc = wmma_f32(a, b, acc);
      }
      const int col = n0 * 16 + mn;
      const float bias = sB2[col];
#pragma unroll
      for (int j = 0; j < 8; ++j)
        H2[(j + mh) * 65 + col] = fmaxf(acc[j] + bias, 0.f);
    }

    // sdf = h2 @ W3 + b3: both half-waves each reduce 32 columns of one row,
    // then combine across the half-waves with a wave32 shuffle.
    {
      float part = (lane < 16) ? sB3[0] : 0.f;
      const int kof = (lane >> 4) * 32;
#pragma unroll
      for (int j = 0; j < 32; ++j)
        part = fmaf(H2[mn * 65 + kof + j], sW3[kof + j], part);
      part += __shfl_xor(part, 16, 32);
      if (lane < 16) out[blockIdx.x * BLK + pb + mn] = part;
    }

    // Backward: g1 = (g2 @ W2^T) * relu'(h1),  g2 = W3 * relu'(h2)
    // g2 A-fragments are synthesized on the fly from h2 mask + W3.
#pragma unroll 1
    for (int n0 = 0; n0 < 4; ++n0) {
      v8f acc = {};
#pragma unroll
      for (int kk = 0; kk < 64; kk += 4) {
        const int k0 = kk + kb, k1 = kk + kb + 1;
        v2f a, b;
        a.x = (H2[mn * 65 + k0] > 0.f) ? sW3[k0] : 0.f;
        a.y = (H2[mn * 65 + k1] > 0.f) ? sW3[k1] : 0.f;
        b.x = sW2[(n0 * 16 + mn) * 64 + k0];  // W2^T
        b.y = sW2[(n0 * 16 + mn) * 64 + k1];
        acc = wmma_f32(a, b, acc);
      }
      const int col = n0 * 16 + mn;
#pragma unroll
      for (int j = 0; j < 8; ++j) {
        const int off = (j + mh) * 65 + col;
        const float hv = H1[off];
        H1[off] = (hv > 0.f) ? acc[j] : 0.f;  // g1 overwrites h1 in place
      }
    }

    // g0 = g1 @ W1^T   (16x64)x(64x32) -> overwrite feature rows with grads
#pragma unroll 1
    for (int n0 = 0; n0 < 2; ++n0) {
      v8f acc = {};
#pragma unroll
      for (int kk = 0; kk < 64; kk += 4) {
        const int k0 = kk + kb, k1 = kk + kb + 1;
        v2f a, b;
        a.x = H1[mn * 65 + k0];
        a.y = H1[mn * 65 + k1];
        b.x = sW1[(n0 * 16 + mn) * 64 + k0];  // W1^T
        b.y = sW1[(n0 * 16 + mn) * 64 + k1];
        acc = wmma_f32(a, b, acc);
      }
      const int col = n0 * 16 + mn;
#pragma unroll
      for (int j = 0; j < 8; ++j) sH0[pb + j + mh][col] = acc[j];
    }
  }
  __syncthreads();

  // ---------------- Phase 3: encoding backward (recompute + re-gather) -----
  float gp0 = 0.f, gp1 = 0.f, gp2 = 0.f, gp3 = 0.f;
#pragma unroll 2
  for (int l = 0; l < LVLS; ++l) {
    const float scale = exp2f(fmaf((float)l, 7.0f / 15.0f, 4.0f));
    int idx[5], rank[5];
    float bary[5];
    permuto_level(pos, scale, idx, rank, bary);
    const float gf0 = sH0[tid][2 * l];
    const float gf1 = sH0[tid][2 * l + 1];
    const float2* tb = (const float2*)table + (size_t)l * TSZ;
    float gb[6];
#pragma unroll
    for (int k = 0; k < 5; ++k) {
      const float2 v = tb[idx[k]];
      gb[k] = fmaf(gf0, v.x, gf1 * v.y);  // dL/dbary_k
    }
    gb[5] = gb[0];  // bary[0] += bary[5] chain
    float ge[5];
#pragma unroll
    for (int v5 = 0; v5 < 5; ++v5) {
      const int bp = 4 - rank[v5];
      const int bm = 5 - rank[v5];
      float gd = 0.f;
#pragma unroll
      for (int b = 0; b < 6; ++b) {
        gd += (bp == b) ? gb[b] : 0.f;
        gd -= (bm == b) ? gb[b] : 0.f;
      }
      ge[v5] = gd * 0.2f;  // d elevated
    }
    const float s01 = ge[0] + ge[1];
    const float gc0 = ge[0] - ge[1];
    const float gc1 = s01 - 2.f * ge[2];
    const float gc2 = s01 + ge[2] - 3.f * ge[3];
    const float gc3 = s01 + ge[2] + ge[3] - 4.f * ge[4];
    gp0 = fmaf(gc0 * scale, SF0, gp0);
    gp1 = fmaf(gc1 * scale, SF1, gp1);
    gp2 = fmaf(gc2 * scale, SF2, gp2);
    gp3 = fmaf(gc3 * scale, SF3, gp3);
  }
  float* nab = out + NPTS;
  nab[(size_t)gp * 4 + 0] = gp0 * 0.5f;
  nab[(size_t)gp * 4 + 1] = gp1 * 0.5f;
  nab[(size_t)gp * 4 + 2] = gp2 * 0.5f;
  nab[(size_t)gp * 4 + 3] = gp3 * 0.5f;
}

extern "C" void kernel_launch(void* const* d_in, const int* in_sizes, int n_in,
                              void* d_out, int out_size, void* d_ws,
                              size_t ws_size, hipStream_t stream) {
  const float* x = (const float*)d_in[0];
  const float* z = (const float*)d_in[1];
  const float* table = (const float*)d_in[2];
  const float* W1 = (const float*)d_in[3];
  const float* b1 = (const float*)d_in[4];
  const float* W2 = (const float*)d_in[5];
  const float* b2 = (const float*)d_in[6];
  const float* W3 = (const float*)d_in[7];
  const float* b3 = (const float*)d_in[8];
  dim3 grid(NPTS / BLK), block(BLK);
  hipLaunchKernelGGL(permuto_sdf_kernel, grid, block, 0, stream, x, z, table,
                     W1, b1, W2, b2, W3, b3, (float*)d_out);
}